// MyMseLoss_82635170775361
// MI455X (gfx1250) — compile-verified
//
#include <hip/hip_runtime.h>

#define NUM_CLASSES   38
#define BLOCK         128                      // 4 waves (wave32)
#define ROWS_PER_TILE 128                      // one row per thread
#define TILE_FLOATS   (ROWS_PER_TILE * NUM_CLASSES)   // 4864 floats = 19456 B
#define TILE_CHUNKS   (TILE_FLOATS / 4)                // 1216 x 16B chunks
#define COPY_ROUNDS   ((TILE_CHUNKS + BLOCK - 1) / BLOCK) // 10 (uniform per wave)
#define MAX_BLOCKS    2048

// ---------------- CDNA5 async + TDM staging -------------------------------

typedef __attribute__((__vector_size__(16))) int v4i;
typedef __attribute__((address_space(1))) v4i* global_v4i_p;  // ptr to v4i in AS1
typedef __attribute__((address_space(3))) v4i* lds_v4i_p;     // ptr to v4i in AS3

typedef unsigned int u32x4 __attribute__((ext_vector_type(4)));
typedef int          i32x8 __attribute__((ext_vector_type(8)));
typedef int          i32x4 __attribute__((ext_vector_type(4)));

#if __has_builtin(__builtin_amdgcn_s_wait_asynccnt)
#define WAIT_ASYNCCNT(n) __builtin_amdgcn_s_wait_asynccnt(n)
#else
#define WAIT_ASYNCCNT(n) asm volatile("s_wait_asynccnt %0" ::"i"(n) : "memory")
#endif

#if __has_builtin(__builtin_amdgcn_s_wait_tensorcnt)
#define WAIT_TENSORCNT(n) __builtin_amdgcn_s_wait_tensorcnt(n)
#else
#define WAIT_TENSORCNT(n) asm volatile("s_wait_tensorcnt %0" ::"i"(n) : "memory")
#endif

#if __has_builtin(__builtin_amdgcn_tensor_load_to_lds)
#define HAVE_TDM 1
#endif

__device__ __forceinline__ void copy16_async(const float* __restrict__ g,
                                             float* l) {
#if __has_builtin(__builtin_amdgcn_global_load_async_to_lds_b128)
  // async DMA: global 16B -> LDS 16B, tracked by ASYNCcnt
  __builtin_amdgcn_global_load_async_to_lds_b128(
      (global_v4i_p)g, (lds_v4i_p)l, /*imm offset*/ 0, /*cpol*/ 0);
#else
  *(float4*)l = *(const float4*)g;   // synchronous fallback
#endif
}

#if defined(HAVE_TDM)
// One TDM descriptor per tile: flat 1D copy of TILE_FLOATS f32 to LDS.
// D# per ISA ch.8: group0 = {count|flags, lds_addr, gaddr lo, gaddr hi|type=2},
// group1 = {data_size=4B, tensor_dim0=4864, tensor_dim1=1, tile_dim0=4864,
//           tile_dim1=1, dim0_stride=4864}, groups 2/3 = 0 (dims unused).
// This toolchain exposes the 6-arg builtin:
//   (u32x4 g0, i32x8 g1, i32x4 g2, i32x4 g3, i32x8 extra, i32 cpol)
__device__ __forceinline__ void tdm_load_tile(const float* __restrict__ preds,
                                              int tile, float* dstLds) {
  const unsigned long long gaddr =
      (unsigned long long)preds + 4ull * (unsigned long long)tile * TILE_FLOATS;
  const unsigned ldsAddr = (unsigned)(unsigned long long)(lds_v4i_p)dstLds;

  u32x4 g0;
  g0.x = 1u;                                                    // count=1 (valid)
  g0.y = ldsAddr;                                               // LDS byte addr
  g0.z = (unsigned)(gaddr & 0xffffffffull);                     // gaddr[31:0]
  g0.w = (unsigned)((gaddr >> 32) & 0x1ffffffull) | (2u << 30); // gaddr[56:32]|type=2

  i32x8 g1;
  g1[0] = 0x00020000;                            // wg_mask=0, data_size=2 (4B)
  g1[1] = (int)((TILE_FLOATS & 0xffff) << 16);   // tensor_dim0[15:0]
  g1[2] = (int)(((TILE_FLOATS >> 16) & 0xffff) | (1u << 16)); // dim0 hi | tensor_dim1=1
  g1[3] = (int)((TILE_FLOATS & 0xffff) << 16);   // tensor_dim1 hi=0 | tile_dim0
  g1[4] = 1;                                     // tile_dim1=1, tile_dim2=0
  g1[5] = TILE_FLOATS;                           // tensor_dim0_stride[31:0]
  g1[6] = 0;                                     // stride hi, dim1_stride lo
  g1[7] = 0;

  const i32x4 gz4 = {0, 0, 0, 0};
  const i32x8 gz8 = {0, 0, 0, 0, 0, 0, 0, 0};
  __builtin_amdgcn_tensor_load_to_lds(g0, g1, gz4, gz4, gz8, /*cpol*/ 0);
}
#else
// Fallback tile staging: async b128 per lane, exactly COPY_ROUNDS per wave.
__device__ __forceinline__ void prefetch_tile(const float* __restrict__ preds,
                                              int tile, float* dst) {
  const int tid = threadIdx.x;
  const long long baseF = (long long)tile * TILE_FLOATS;  // 16B-aligned
#pragma unroll
  for (int k = 0; k < COPY_ROUNDS; ++k) {
    int c = tid + k * BLOCK;
    c = (c < TILE_CHUNKS) ? c : (TILE_CHUNKS - 1);  // uniform wave issue count
    copy16_async(preds + baseF + 4LL * (long long)c, dst + 4 * c);
  }
}
#endif

// 128 labels = 512 B = exactly 32 x 16B chunks: one async b128 per lane of
// wave 0 -> ASYNCcnt contribution of exactly 1 per prefetch for wave 0.
__device__ __forceinline__ void prefetch_labels(const int* __restrict__ labels,
                                                int tile, int B, int* dst) {
  const int lane = threadIdx.x;  // called by wave 0 only: 0..31
  long long base = (long long)tile * ROWS_PER_TILE + 4LL * lane;
  long long maxb = (long long)B - 4;
  if (maxb < 0) maxb = 0;
  if (base > maxb) base = maxb;  // tail rows are masked in compute anyway
  copy16_async((const float*)(labels + base), (float*)(dst + 4 * lane));
}

__device__ __forceinline__ void stage_tile(const float* __restrict__ preds,
                                           const int* __restrict__ labels,
                                           int tile, int B, float* tbuf,
                                           int* lbuf, bool wave0) {
#if defined(HAVE_TDM)
  if (wave0) tdm_load_tile(preds, tile, tbuf);   // 1 TDM op, TENSORcnt
#else
  prefetch_tile(preds, tile, tbuf);              // async b128, ASYNCcnt
#endif
  if (wave0) prefetch_labels(labels, tile, B, lbuf);
}

// ---------------- main loss kernel ----------------------------------------

__global__ void __launch_bounds__(BLOCK)
ordinal_loss_kernel(const float* __restrict__ preds,
                    const int* __restrict__ labels,
                    const float* __restrict__ lossmat,
                    float* __restrict__ partials,
                    int B, int tiles) {
  __shared__ __align__(16) float sbuf[2][TILE_FLOATS];  // double-buffered tiles
  __shared__ __align__(16) int   slab[2][ROWS_PER_TILE];
  __shared__ float sM[NUM_CLASSES * NUM_CLASSES];
  __shared__ float sred[BLOCK];

  const int tid = threadIdx.x;
  const bool wave0 = (tid < 32);  // uniform per wave

  // stage the 38x38 loss matrix once
  for (int i = tid; i < NUM_CLASSES * NUM_CLASSES; i += BLOCK)
    sM[i] = lossmat[i];

  float acc = 0.0f;

  int t = blockIdx.x;
  if (t < tiles) stage_tile(preds, labels, t, B, sbuf[0], slab[0], wave0);
  int parity = 0;

  for (t = blockIdx.x; t < tiles; t += gridDim.x) {
    const int tn = t + gridDim.x;
    if (tn < tiles) {
      stage_tile(preds, labels, tn, B, sbuf[parity ^ 1], slab[parity ^ 1], wave0);
#if defined(HAVE_TDM)
      WAIT_ASYNCCNT(1);    // wave0: labels for tile t landed (in-order)
      WAIT_TENSORCNT(1);   // wave0: TDM for tile t complete; others trivial
#else
      WAIT_ASYNCCNT(COPY_ROUNDS);  // previous batch complete (over-wait safe)
#endif
    } else {
      WAIT_ASYNCCNT(0);
#if defined(HAVE_TDM)
      WAIT_TENSORCNT(0);
#endif
    }
    __syncthreads();  // tile t (and sM on first pass) visible block-wide

    const int row = t * ROWS_PER_TILE + tid;
    if (row < B) {
      const float* xr = &sbuf[parity][tid * NUM_CLASSES];
      float x[NUM_CLASSES];
#pragma unroll
      for (int j = 0; j < NUM_CLASSES; ++j) x[j] = xr[j];

      float mx = x[0];
#pragma unroll
      for (int j = 1; j < NUM_CLASSES; ++j) mx = fmaxf(mx, x[j]);

      const int lab = slab[parity][tid];
      const float* Mr = &sM[lab * NUM_CLASSES];

      float s = 0.0f, d = 0.0f;
#pragma unroll
      for (int j = 0; j < NUM_CLASSES; ++j) {
        const float e = __expf(x[j] - mx);  // v_exp_f32 (TRANS, co-issues)
        s += e;
        d = fmaf(e, Mr[j], d);
      }
      acc += d / s;
    }
    __syncthreads();  // all reads done before buffers are re-filled
    parity ^= 1;
  }

  // deterministic block tree-reduction
  sred[tid] = acc;
  __syncthreads();
#pragma unroll
  for (int off = BLOCK / 2; off > 0; off >>= 1) {
    if (tid < off) sred[tid] += sred[tid + off];
    __syncthreads();
  }
  if (tid == 0) partials[blockIdx.x] = sred[0];
}

// ---------------- final deterministic reduce ------------------------------

__global__ void __launch_bounds__(256)
final_reduce_kernel(const float* __restrict__ partials, int n, int B,
                    float* __restrict__ out) {
  __shared__ double sred[256];
  const int tid = threadIdx.x;
  double a = 0.0;
  for (int i = tid; i < n; i += 256) a += (double)partials[i];
  sred[tid] = a;
  __syncthreads();
  for (int off = 128; off > 0; off >>= 1) {
    if (tid < off) sred[tid] += sred[tid + off];
    __syncthreads();
  }
  if (tid == 0) out[0] = (float)(sred[0] / (double)B);
}

// ---------------- launch ---------------------------------------------------

extern "C" void kernel_launch(void* const* d_in, const int* in_sizes, int n_in,
                              void* d_out, int out_size, void* d_ws,
                              size_t ws_size, hipStream_t stream) {
  const float* preds  = (const float*)d_in[0];   // [B, 38] f32
  const int*   labels = (const int*)d_in[1];     // [B] int32 (JAX x64 off)
  const float* lossm  = (const float*)d_in[2];   // [38, 38] f32
  float* out = (float*)d_out;

  const int B = in_sizes[1];
  const int tiles = (B + ROWS_PER_TILE - 1) / ROWS_PER_TILE;
  const int nblocks = tiles < MAX_BLOCKS ? tiles : MAX_BLOCKS;

  float* partials = (float*)d_ws;  // nblocks floats of scratch

  hipLaunchKernelGGL(ordinal_loss_kernel, dim3(nblocks), dim3(BLOCK), 0, stream,
                     preds, labels, lossm, partials, B, tiles);
  hipLaunchKernelGGL(final_reduce_kernel, dim3(1), dim3(256), 0, stream,
                     partials, nblocks, B, out);
}